// FeatureMemory_67654324847624
// MI455X (gfx1250) — compile-verified
//
#include <hip/hip_runtime.h>
#include <stdint.h>

#define NUM_CLASSES 124
#define FEAT_CH     64
#define OUT_HW      480
#define IN_HW       120
#define NPIX        (OUT_HW*OUT_HW)
#define EPSV        1e-8f

// ---------- workspace float layout ----------
#define WS_MN    0                                  // 124*64 normalized memory
#define WS_FLAG  (WS_MN + NUM_CLASSES*FEAT_CH)      // 124 is_zero flags
#define WS_ACC   (WS_FLAG + 128)                    // 124*66: [count, denom, acc0..63]
#define ACC_G    66
#define ACC_G_TOT (NUM_CLASSES*ACC_G)

// ---------- LDS float layout (dynamic) ----------
#define L_MN        0
#define L_MN_STR    65                              // odd stride -> conflict-free gathers
#define L_ACC       (NUM_CLASSES*L_MN_STR)          // 8060
#define L_ACC_STR   67                              // odd stride -> spread atomics
#define LDS_FLOATS  (L_ACC + NUM_CLASSES*L_ACC_STR) // 16368 floats = 65472 B

// -------- zero the global accumulators (fresh every launch) --------
__global__ void __launch_bounds__(256) k_zero(float* __restrict__ gAcc) {
    int i = blockIdx.x * 256 + threadIdx.x;
    if (i < ACC_G_TOT) gAcc[i] = 0.0f;
}

// -------- normalize memory rows, detect all-zero rows (1 wave / class) --------
__global__ void __launch_bounds__(32) k_prep(const float* __restrict__ mem,
                                             float* __restrict__ gMn,
                                             float* __restrict__ gFlag) {
    int c = blockIdx.x, l = threadIdx.x;
    float a = mem[c*FEAT_CH + l];
    float b = mem[c*FEAT_CH + 32 + l];
    float nsq = a*a + b*b;
    int   zf  = (a == 0.0f) && (b == 0.0f);
    #pragma unroll
    for (int off = 16; off >= 1; off >>= 1) {      // wave32 butterfly reduce
        nsq += __shfl_xor(nsq, off, 32);
        zf  &= __shfl_xor(zf,  off, 32);
    }
    float inv = 1.0f / fmaxf(sqrtf(nsq), EPSV);    // zero row -> mn = 0 -> sim = 0 -> w = 1
    gMn[c*FEAT_CH + l]      = a * inv;
    gMn[c*FEAT_CH + 32 + l] = b * inv;
    if (l == 0) gFlag[c] = zf ? 1.0f : 0.0f;
}

// -------- main pass: bilinear upsample + per-class scatter into LDS --------
__global__ void __launch_bounds__(256) k_main(const float* __restrict__ feats,
                                              const int*   __restrict__ seg,
                                              const float* __restrict__ gMn,
                                              float*       __restrict__ gAcc) {
    extern __shared__ float smem[];
    const int tid = threadIdx.x;

    // zero LDS accumulators
    for (int i = tid; i < NUM_CLASSES*L_ACC_STR; i += 256) smem[L_ACC + i] = 0.0f;

    // async global->LDS copy of normalized memory, re-padded to stride 65
    // (CDNA5 GLOBAL_LOAD_ASYNC_TO_LDS, tracked by ASYNCcnt)
    uint32_t ldsbase = __builtin_amdgcn_groupstaticsize();
    for (int i = tid; i < NUM_CLASSES*FEAT_CH; i += 256) {   // 7936 = 31*256, uniform
        int r = i >> 6, c = i & 63;
        uint32_t dst = ldsbase + (uint32_t)(L_MN + r*L_MN_STR + c) * 4u;
        uint64_t src = (uint64_t)(uintptr_t)(gMn + i);
        asm volatile("global_load_async_to_lds_b32 %0, %1, off"
                     :: "v"(dst), "v"(src) : "memory");
    }
    asm volatile("s_wait_asynccnt 0x0" ::: "memory");
    __syncthreads();

    const int stride = gridDim.x * blockDim.x;
    for (int p = blockIdx.x * blockDim.x + tid; p < NPIX; p += stride) {
        int oy = p / OUT_HW, ox = p - oy * OUT_HW;
        // jax.image.resize bilinear (upsample x4): t = (o+0.5)/4 - 0.5, clamped taps
        float ty = (oy + 0.5f) * 0.25f - 0.5f;
        float tx = (ox + 0.5f) * 0.25f - 0.5f;
        float fy0 = floorf(ty), fx0 = floorf(tx);
        float fy = ty - fy0,    fx = tx - fx0;
        int y0 = (int)fy0, x0 = (int)fx0;
        int y1 = min(y0 + 1, IN_HW - 1); y0 = max(y0, 0);
        int x1 = min(x0 + 1, IN_HW - 1); x0 = max(x0, 0);
        float w00 = (1.f-fy)*(1.f-fx), w01 = (1.f-fy)*fx;
        float w10 = fy*(1.f-fx),       w11 = fy*fx;
        int r0 = y0 * IN_HW, r1 = y1 * IN_HW;

        int cls = seg[p];
        const float* mrow = &smem[L_MN + cls * L_MN_STR];

        float xv[FEAT_CH];
        float nsq = 0.f, dotm = 0.f;
        #pragma unroll
        for (int k = 0; k < FEAT_CH; ++k) {
            const float* f = feats + k * (IN_HW * IN_HW);
            float v = w00*f[r0+x0] + w01*f[r0+x1] + w10*f[r1+x0] + w11*f[r1+x1];
            xv[k] = v;
            nsq  = fmaf(v, v, nsq);
            dotm = fmaf(v, mrow[k], dotm);
        }
        float sim = dotm / fmaxf(sqrtf(nsq), EPSV);
        float w = 1.0f - sim;                       // zero-mem classes: sim==0 -> w==1

        float* arow = &smem[L_ACC + cls * L_ACC_STR];
        atomicAdd(&arow[0], 1.0f);                  // count
        atomicAdd(&arow[1], w);                     // denom
        #pragma unroll
        for (int k = 0; k < FEAT_CH; ++k)
            atomicAdd(&arow[2 + k], w * xv[k]);     // ds_add_f32, odd stride
    }
    __syncthreads();

    // flush block-private accumulators to global
    for (int i = tid; i < ACC_G_TOT; i += 256) {
        int cls = i / ACC_G, slot = i - cls * ACC_G;
        float v = smem[L_ACC + cls * L_ACC_STR + slot];
        if (v != 0.0f) atomicAdd(&gAcc[i], v);
    }
}

// -------- finalize: momentum blend / mean / passthrough --------
__global__ void __launch_bounds__(256) k_final(const float* __restrict__ mem,
                                               const float* __restrict__ gFlag,
                                               const float* __restrict__ gAcc,
                                               float*       __restrict__ out) {
    int i = blockIdx.x * 256 + threadIdx.x;        // 7936 = 31*256
    int cls = i >> 6;
    float count = gAcc[cls * ACC_G + 0];
    float denom = gAcc[cls * ACC_G + 1];
    float a     = gAcc[cls * ACC_G + 2 + (i & 63)];
    float m     = mem[i];
    float weighted = a / ((denom != 0.0f) ? denom : 1.0f);
    float meanv    = a / fmaxf(count, 1.0f);
    float upd      = 0.9f * m + 0.1f * weighted;
    float newv     = (gFlag[cls] != 0.0f) ? meanv : upd;
    out[i] = (count > 0.0f) ? newv : m;
}

extern "C" void kernel_launch(void* const* d_in, const int* in_sizes, int n_in,
                              void* d_out, int out_size, void* d_ws, size_t ws_size,
                              hipStream_t stream) {
    (void)in_sizes; (void)n_in; (void)out_size; (void)ws_size;
    const float* feats = (const float*)d_in[0];     // (1,64,120,120)
    const float* mem   = (const float*)d_in[1];     // (124,1,64)
    const int*   seg   = (const int*)  d_in[2];     // (1,480,480) int32
    float* ws   = (float*)d_ws;
    float* gMn  = ws + WS_MN;
    float* gFlg = ws + WS_FLAG;
    float* gAcc = ws + WS_ACC;
    float* out  = (float*)d_out;

    k_zero <<<(ACC_G_TOT + 255)/256, 256, 0, stream>>>(gAcc);
    k_prep <<<NUM_CLASSES, 32, 0, stream>>>(mem, gMn, gFlg);
    k_main <<<225, 256, LDS_FLOATS * sizeof(float), stream>>>(feats, seg, gMn, gAcc);
    k_final<<<NUM_CLASSES*FEAT_CH/256, 256, 0, stream>>>(mem, gFlg, gAcc, out);
}